// WaveletInterpolate_54855322304806
// MI455X (gfx1250) — compile-verified
//
#include <hip/hip_runtime.h>
#include <cstdint>
#include <cstddef>

// WaveletInterpolate fused kernel for MI455X (gfx1250).
// out(b, 4 bands, 256, 256) computed directly from x(b, 256, 128, 128):
//   IDWT -> bilinear 2x upsample -> DWT collapses to a 3x3 stencil over the
//   (never materialized) 256x256 IDWT image, which itself is a parity butterfly
//   of the 4 input subbands. Memory-bound: ~671 MB min traffic ~= 29us roofline.
//
// CDNA5 paths used:
//   - global_load_async_to_lds_b32 (ASYNCcnt) to stage subband tiles in LDS
//   - s_wait_asynccnt before the workgroup barrier
//   - non-temporal b128 stores for the 537MB streaming output (preserve L2
//     for the input halo reuse; output is never re-read)

#define CIN   256   // input channels
#define CC    64    // wavelet channels
#define HS    128   // subband spatial size (input)
#define HO    256   // output band spatial size
#define TILE  32    // output tile edge per block
#define SB_R  18    // subband tile rows/cols needed (TILE/2 + 2)
#define SB_S  19    // padded LDS stride for subband tile
#define IM_R  34    // img tile rows/cols needed (TILE + 2)
#define IM_S  35    // padded LDS stride for img tile

typedef __attribute__((ext_vector_type(4))) float v4f;

static __device__ __forceinline__ int clampi(int v, int lo, int hi) {
    return v < lo ? lo : (v > hi ? hi : v);
}

__global__ __launch_bounds__(256)
void wavelet_interp_fused(const float* __restrict__ x, float* __restrict__ out)
{
    const int tid   = threadIdx.x;
    const int tileq = blockIdx.x & 7;
    const int tilep = blockIdx.x >> 3;
    const int c     = blockIdx.y;   // wavelet channel 0..63
    const int b     = blockIdx.z;   // batch 0..7

    const int p0 = tilep * TILE;
    const int q0 = tileq * TILE;
    const int i0 = (p0 >> 1) - 1;   // subband tile row origin (may be -1)
    const int j0 = (q0 >> 1) - 1;

    __shared__ float sB[4 * SB_R * SB_S];   // raw subband tiles (ll,lh,hl,hh)
    __shared__ float img[IM_R * IM_S];      // IDWT image tile, rows p0-1..p0+32

    const size_t planeIn = (size_t)HS * HS;
    const size_t base0 = ((size_t)b * CIN + c) * planeIn;                  // ll
    const size_t base1 = ((size_t)b * CIN + (CC + 3 * c + 0)) * planeIn;   // lh
    const size_t base2 = ((size_t)b * CIN + (CC + 3 * c + 1)) * planeIn;   // hl
    const size_t base3 = ((size_t)b * CIN + (CC + 3 * c + 2)) * planeIn;   // hh

    // ---------------- Stage 0: async DMA global -> LDS (ASYNCcnt path) -----
    const int NSB = 4 * SB_R * SB_R;  // 1296 dwords
    for (int f = tid; f < NSB; f += 256) {
        int k  = f / (SB_R * SB_R);
        int rm = f - k * (SB_R * SB_R);
        int sr = rm / SB_R;
        int sc = rm - sr * SB_R;
        int gi = clampi(i0 + sr, 0, HS - 1);   // edge-clamp at load time
        int gj = clampi(j0 + sc, 0, HS - 1);
        size_t base = (k == 0) ? base0 : (k == 1) ? base1 : (k == 2) ? base2 : base3;
        const float* gp = x + base + (size_t)gi * HS + gj;
        unsigned lds_off = (unsigned)(uintptr_t)(&sB[(k * SB_R + sr) * SB_S + sc]);
        // per-lane async copy: memory -> LDS, no VGPR data, tracked by ASYNCcnt
        asm volatile("global_load_async_to_lds_b32 %0, %1, off"
                     :: "v"(lds_off), "v"(gp)
                     : "memory");
    }
#if __has_builtin(__builtin_amdgcn_s_wait_asynccnt)
    __builtin_amdgcn_s_wait_asynccnt(0);
#else
    asm volatile("s_wait_asynccnt 0" ::: "memory");
#endif
    __syncthreads();

    // ---------------- Stage 1: IDWT butterfly into img tile ----------------
    // img global row r maps to subband row r>>1; parity selects a/b/c/d:
    //   val = 0.5*(ll + s_r*lh + s_c*hl + s_r*s_c*hh),  s = +1 even / -1 odd
    const int NIMG = IM_R * IM_R;  // 1156
    for (int e = tid; e < NIMG; e += 256) {
        int lr = e / IM_R;
        int lc = e - lr * IM_R;
        int r  = clampi(p0 - 1 + lr, 0, HO - 1);
        int g  = clampi(q0 - 1 + lc, 0, HO - 1);
        int si = (r >> 1) - i0;
        int sj = (g >> 1) - j0;
        float ll = sB[(0 * SB_R + si) * SB_S + sj];
        float lh = sB[(1 * SB_R + si) * SB_S + sj];
        float hl = sB[(2 * SB_R + si) * SB_S + sj];
        float hh = sB[(3 * SB_R + si) * SB_S + sj];
        float sr_ = (r & 1) ? -1.0f : 1.0f;
        float sc_ = (g & 1) ? -1.0f : 1.0f;
        img[lr * IM_S + lc] = 0.5f * (ll + sr_ * lh + sc_ * hl + (sr_ * sc_) * hh);
    }
    __syncthreads();

    // ---------------- Stage 2: upsample stencil + DWT, NT b128 stores ------
    // Thread -> (row tp, column quad qd): 32 rows x 8 quads = 256 threads.
    const int qd  = tid & 7;
    const int tp  = tid >> 3;
    const int lc0 = 4 * qd;            // img local col base (covers g = q-1..q+4)

    const size_t planeOut = (size_t)HO * HO;
    float* oLL = out + ((size_t)b * CIN + c) * planeOut;
    float* oLH = out + ((size_t)b * CIN + (CC + 3 * c + 0)) * planeOut;
    float* oHL = out + ((size_t)b * CIN + (CC + 3 * c + 1)) * planeOut;
    float* oHH = out + ((size_t)b * CIN + (CC + 3 * c + 2)) * planeOut;

    // three img rows (p-1, p, p+1), six columns each
    float v0[6], v1[6], v2[6];
    #pragma unroll
    for (int u = 0; u < 6; ++u) {
        v0[u] = img[(tp + 0) * IM_S + lc0 + u];
        v1[u] = img[(tp + 1) * IM_S + lc0 + u];
        v2[u] = img[(tp + 2) * IM_S + lc0 + u];
    }

    v4f rLL, rLH, rHL, rHH;

    #pragma unroll
    for (int u = 0; u < 4; ++u) {
        // horizontal upsample taps per row: even col (A) and odd col (B)
        float hA0 = 0.25f * v0[u] + 0.75f * v0[u + 1];
        float hB0 = 0.75f * v0[u + 1] + 0.25f * v0[u + 2];
        float hA1 = 0.25f * v1[u] + 0.75f * v1[u + 1];
        float hB1 = 0.75f * v1[u + 1] + 0.25f * v1[u + 2];
        float hA2 = 0.25f * v2[u] + 0.75f * v2[u + 1];
        float hB2 = 0.75f * v2[u + 1] + 0.25f * v2[u + 2];
        // vertical upsample: 2x2 upsampled block (A B / C D)
        float A  = 0.25f * hA0 + 0.75f * hA1;
        float Bv = 0.25f * hB0 + 0.75f * hB1;
        float Cv = 0.75f * hA1 + 0.25f * hA2;
        float Dv = 0.75f * hB1 + 0.25f * hB2;
        // DWT butterfly
        float s1 = A + Bv, s2 = Cv + Dv;
        float d1 = A - Bv, d2 = Cv - Dv;
        rLL[u] = 0.5f * (s1 + s2);
        rLH[u] = 0.5f * (s1 - s2);
        rHL[u] = 0.5f * (d1 + d2);
        rHH[u] = 0.5f * (d1 - d2);
    }

    const size_t o = (size_t)(p0 + tp) * HO + (size_t)(q0 + lc0);
    // streaming output: non-temporal 128-bit stores (th:TH_STORE_NT),
    // coalesced per 8-lane group
    __builtin_nontemporal_store(rLL, (v4f*)(oLL + o));
    __builtin_nontemporal_store(rLH, (v4f*)(oLH + o));
    __builtin_nontemporal_store(rHL, (v4f*)(oHL + o));
    __builtin_nontemporal_store(rHH, (v4f*)(oHH + o));
}

extern "C" void kernel_launch(void* const* d_in, const int* in_sizes, int n_in,
                              void* d_out, int out_size, void* d_ws, size_t ws_size,
                              hipStream_t stream) {
    (void)in_sizes; (void)n_in; (void)out_size; (void)d_ws; (void)ws_size;
    const float* x = (const float*)d_in[0];
    float* out = (float*)d_out;
    // 64 tiles per plane (8x8), 64 channels, 8 batches
    dim3 grid(64, 64, 8);
    dim3 block(256);
    wavelet_interp_fused<<<grid, block, 0, stream>>>(x, out);
}